// SlotAttention_86002425135212
// MI455X (gfx1250) — compile-verified
//
#include <hip/hip_runtime.h>
#include <hip/hip_bf16.h>

typedef __attribute__((ext_vector_type(2))) float v2f;
typedef __attribute__((ext_vector_type(8))) float v8f;
typedef __attribute__((ext_vector_type(4))) unsigned u32x4;
typedef __attribute__((ext_vector_type(4))) int i32x4;
typedef __attribute__((ext_vector_type(8))) int i32x8;

#define TILE_ROWS 64
#define LDS_STRIDE 260   // 256 + 4 pad (floats); TDM pad_interval=7/pad_amount=3 reproduces this
#define SEQ 4096
#define DIN 256
#define DSL 64
#define NSLOT 8
#define NB 32
#define TOTAL_ROWS (NB * SEQ)

#if defined(__gfx1250__) && __has_builtin(__builtin_amdgcn_tensor_load_to_lds)
#define USE_TDM 1
#endif

// ---------------------------------------------------------------------------
// Kernel 0: broadcast slot_emb (1,K,D) -> slots (B,K,D)
// ---------------------------------------------------------------------------
__global__ void sa_init_slots(const float* __restrict__ se, float* __restrict__ slots) {
    int i = blockIdx.x * 256 + threadIdx.x;          // 16384 total
    slots[i] = se[i & (NSLOT * DSL - 1)];
}

// ---------------------------------------------------------------------------
// Kernel 1: fused LayerNorm(inputs) + [k|v] projection, f32 WMMA 16x16x4.
// grid = (B*S)/64 blocks, 256 threads (8 wave32).
// Input tile staged by the Tensor Data Mover (async DMA, no VGPR transit),
// with TDM LDS-padding producing the 260-float row stride.
// ---------------------------------------------------------------------------
__global__ __launch_bounds__(256) void sa_ln_kv(
    const float* __restrict__ inputs,
    const float* __restrict__ Wk, const float* __restrict__ Wv,
    const float* __restrict__ lng, const float* __restrict__ lnb,
    float* __restrict__ kout, float* __restrict__ vout)
{
    __shared__ float sIn[TILE_ROWS * LDS_STRIDE];

    const int tid  = threadIdx.x;
    const int lane = tid & 31;
    const int wave = tid >> 5;
    const long long row0 = (long long)blockIdx.x * TILE_ROWS;

    // warm L2 with the (tiny, fully reused) weight matrices
    __builtin_prefetch(Wk + (size_t)tid * 64, 0, 1);
    __builtin_prefetch(Wv + (size_t)tid * 64, 0, 1);

#ifdef USE_TDM
    // ---- stage 64x256 f32 tile into LDS via Tensor Data Mover ----
    if (wave == 0) {
        const unsigned lds_off = (unsigned)(size_t)(&sIn[0]);   // low 32b = LDS byte offset
        const unsigned long long ga =
            (unsigned long long)(const void*)(inputs + row0 * DIN);
        const unsigned rem = (unsigned)(TOTAL_ROWS - row0);     // rows left (OOB bound)

        u32x4 g0;
        g0.x = 1u;                                   // count=1, is_restore=0, gather=0
        g0.y = lds_off;                              // D#.lds_addr (bytes)
        g0.z = (unsigned)(ga & 0xFFFFFFFFu);         // global_addr[31:0]
        g0.w = (unsigned)((ga >> 32) & 0x1FFFFFFu)   // global_addr[56:32]
             | (2u << 30);                           // type = 2 ("image")

        i32x8 g1;
        g1[0] = (int)((2u << 16)     // data_size = 2 -> 4 bytes
                    | (1u << 20)     // pad_enable
                    | (7u << 22)     // pad_interval: every 256 DWORDs (one row)
                    | (3u << 25));   // pad_amount: 4 DWORDs  -> row stride 260 floats
        g1[1] = (int)((DIN & 0xFFFFu) << 16);                    // tensor_dim0 lo16
        g1[2] = (int)(((DIN >> 16) & 0xFFFFu) | ((rem & 0xFFFFu) << 16)); // dim0 hi | dim1 lo
        g1[3] = (int)(((rem >> 16) & 0xFFFFu) | ((unsigned)DIN << 16));   // dim1 hi | tile_dim0
        g1[4] = (int)(TILE_ROWS);                                // tile_dim1 | tile_dim2=0
        g1[5] = (int)(DIN);                                      // tensor_dim0_stride lo32
        g1[6] = 0;
        g1[7] = 0;

        i32x4 gz = {0, 0, 0, 0};                     // VADDR2/3 groups (2-D tensor)
#if __clang_major__ >= 23
        i32x8 gz8 = {0, 0, 0, 0, 0, 0, 0, 0};
        __builtin_amdgcn_tensor_load_to_lds(g0, g1, gz, gz, gz8, 0);
#else
        __builtin_amdgcn_tensor_load_to_lds(g0, g1, gz, gz, 0);
#endif
        __builtin_amdgcn_s_wait_tensorcnt(0);
    }
    __syncthreads();
#else
    // fallback path (host pass / toolchains without the TDM builtin)
    for (int idx = tid; idx < TILE_ROWS * 64; idx += 256) {
        int r  = idx >> 6;
        int c4 = idx & 63;
        float4 v = ((const float4*)(inputs + (row0 + r) * DIN))[c4];
        *(float4*)(&sIn[r * LDS_STRIDE + c4 * 4]) = v;
    }
    __syncthreads();
#endif

    // ---- layernorm in-place: each wave handles 8 rows ----
    for (int r = wave; r < TILE_ROWS; r += 8) {
        float s = 0.f;
        for (int c = lane; c < DIN; c += 32) s += sIn[r * LDS_STRIDE + c];
        for (int off = 16; off > 0; off >>= 1) s += __shfl_xor(s, off, 32);
        float mean = s * (1.f / 256.f);
        float vs = 0.f;
        for (int c = lane; c < DIN; c += 32) {
            float d = sIn[r * LDS_STRIDE + c] - mean;
            vs += d * d;
        }
        for (int off = 16; off > 0; off >>= 1) vs += __shfl_xor(vs, off, 32);
        float rstd = rsqrtf(vs * (1.f / 256.f) + 1e-5f);
        for (int c = lane; c < DIN; c += 32) {
            sIn[r * LDS_STRIDE + c] =
                (sIn[r * LDS_STRIDE + c] - mean) * rstd * lng[c] + lnb[c];
        }
    }
    __syncthreads();

    // ---- GEMM: out(64 x 128) = sIn(64 x 256) @ [Wk|Wv](256 x 128) ----
    const int n    = (wave << 4) | (lane & 15);   // global col 0..127
    const int kb   = (lane >> 4) << 1;            // K offset within a x4 step: 0 or 2
    const int mrow = lane & 15;                   // A-matrix row within tile
    const float* __restrict__ Wcol = (n < DSL) ? Wk : Wv;
    const int ncol = (n < DSL) ? n : (n - DSL);

    v8f acc[4];
    #pragma unroll
    for (int mt = 0; mt < 4; ++mt)
        #pragma unroll
        for (int r = 0; r < 8; ++r) acc[mt][r] = 0.0f;

    for (int kc = 0; kc < DIN; kc += 64) {        // 4 K-chunks
        v2f Bf[16];
        #pragma unroll
        for (int s = 0; s < 16; ++s) {            // B frags: W[k..k+1][ncol]
            int k = kc + s * 4 + kb;
            Bf[s].x = Wcol[(size_t)k * DSL + ncol];
            Bf[s].y = Wcol[(size_t)(k + 1) * DSL + ncol];
        }
        #pragma unroll
        for (int mt = 0; mt < 4; ++mt) {
            #pragma unroll
            for (int s = 0; s < 16; ++s) {
                int k = kc + s * 4 + kb;
                v2f Af;
                Af.x = sIn[(mt * 16 + mrow) * LDS_STRIDE + k];
                Af.y = sIn[(mt * 16 + mrow) * LDS_STRIDE + k + 1];
                // D = A(16x4,f32) * B(4x16,f32) + C(16x16,f32)
                acc[mt] = __builtin_amdgcn_wmma_f32_16x16x4_f32(
                    false, Af, false, Bf[s], (short)0, acc[mt], false, false);
            }
        }
    }

    // ---- store C: vgpr r -> M = r + (lane/16)*8, N = lane%16 ----
    float* __restrict__ dst = (n < DSL) ? kout : vout;
    #pragma unroll
    for (int mt = 0; mt < 4; ++mt) {
        #pragma unroll
        for (int r = 0; r < 8; ++r) {
            int m = mt * 16 + r + ((lane >> 4) << 3);
            dst[(row0 + m) * DSL + ncol] = acc[mt][r];
        }
    }
}

// ---------------------------------------------------------------------------
// Kernel 2: slots layernorm + q = slots_norm @ Wq.  1 block, 256 threads,
// one thread per (b,k) row.
// ---------------------------------------------------------------------------
__global__ __launch_bounds__(256) void sa_slot_q(
    const float* __restrict__ slots, const float* __restrict__ Wq,
    const float* __restrict__ g, const float* __restrict__ b,
    float* __restrict__ qout)
{
    int row = threadIdx.x;                 // 0..255 == B*K
    float x[DSL];
    const float* sr = slots + row * DSL;
    float m = 0.f;
    #pragma unroll
    for (int d = 0; d < DSL; ++d) { x[d] = sr[d]; m += x[d]; }
    m *= (1.f / 64.f);
    float v = 0.f;
    #pragma unroll
    for (int d = 0; d < DSL; ++d) { float t = x[d] - m; v += t * t; }
    float rstd = rsqrtf(v * (1.f / 64.f) + 1e-5f);
    #pragma unroll
    for (int d = 0; d < DSL; ++d) x[d] = (x[d] - m) * rstd * g[d] + b[d];

    for (int j = 0; j < DSL; ++j) {        // Wq loads are lane-uniform (broadcast)
        float a = 0.f;
        #pragma unroll
        for (int d = 0; d < DSL; ++d) a += x[d] * Wq[d * DSL + j];
        qout[row * DSL + j] = a;
    }
}

// ---------------------------------------------------------------------------
// Kernel 3: per-batch scores + softmax + updates (k/v are L2-resident).
// grid = 32 (one block per b), 256 threads; full 8x4096 score matrix in LDS.
// ---------------------------------------------------------------------------
__global__ __launch_bounds__(256) void sa_attn(
    const float* __restrict__ qbuf, const float* __restrict__ kbuf,
    const float* __restrict__ vbuf, const int* __restrict__ mask,
    float* __restrict__ upd, float* __restrict__ attn_out, int write_attn)
{
    __shared__ float qs[NSLOT * DSL];           //   2 KB
    __shared__ float sc[NSLOT][SEQ + 4];        // 131 KB (of 320 KB/WGP)
    __shared__ float part[NSLOT][4][DSL];       //   8 KB

    const int b = blockIdx.x, tid = threadIdx.x;
    const int lane = tid & 31, wave = tid >> 5;

    for (int i = tid; i < NSLOT * DSL; i += 256) qs[i] = qbuf[b * NSLOT * DSL + i];
    __syncthreads();

    // scores = q @ k^T * scale, with mask
    const float* kb_ = kbuf + (size_t)b * SEQ * DSL;
    for (int s = tid; s < SEQ; s += 256) {
        float acc[NSLOT] = {};
        const float4* krow = (const float4*)(kb_ + (size_t)s * DSL);
        #pragma unroll
        for (int c4 = 0; c4 < 16; ++c4) {
            float4 kv = krow[c4];
            #pragma unroll
            for (int kk = 0; kk < NSLOT; ++kk) {
                const float* qr = qs + kk * DSL + c4 * 4;
                acc[kk] += qr[0] * kv.x + qr[1] * kv.y + qr[2] * kv.z + qr[3] * kv.w;
            }
        }
        const bool masked = (mask[b * SEQ + s] == 0);
        #pragma unroll
        for (int kk = 0; kk < NSLOT; ++kk)
            sc[kk][s] = masked ? -3.402823466e38f : acc[kk] * 0.125f;
    }
    __syncthreads();

    // softmax over S: one wave per slot row (8 waves, 8 rows)
    {
        const int kk = wave;
        float mx = -3.402823466e38f;
        for (int s = lane; s < SEQ; s += 32) mx = fmaxf(mx, sc[kk][s]);
        for (int off = 16; off > 0; off >>= 1) mx = fmaxf(mx, __shfl_xor(mx, off, 32));
        float sum = 0.f;
        for (int s = lane; s < SEQ; s += 32) {
            float e = __expf(sc[kk][s] - mx);
            sc[kk][s] = e;
            sum += e;
        }
        for (int off = 16; off > 0; off >>= 1) sum += __shfl_xor(sum, off, 32);
        float inv = 1.f / sum;
        for (int s = lane; s < SEQ; s += 32) {
            float a = sc[kk][s] * inv;
            sc[kk][s] = a;
            if (write_attn)
                attn_out[((size_t)(b * NSLOT + kk)) * SEQ + s] = a;
        }
    }
    __syncthreads();

    // updates = attn @ v  (v reads coalesced over d; sc reads broadcast)
    {
        const float* vb_ = vbuf + (size_t)b * SEQ * DSL;
        const int d = tid & 63, gq = tid >> 6;   // 4 s-phases
        float acc[NSLOT] = {};
        for (int s = gq; s < SEQ; s += 4) {
            float vv = vb_[(size_t)s * DSL + d];
            #pragma unroll
            for (int kk = 0; kk < NSLOT; ++kk) acc[kk] += sc[kk][s] * vv;
        }
        #pragma unroll
        for (int kk = 0; kk < NSLOT; ++kk) part[kk][gq][d] = acc[kk];
    }
    __syncthreads();

    for (int i = tid; i < NSLOT * DSL; i += 256) {
        int kk = i >> 6, d = i & 63;
        upd[((size_t)b * NSLOT + kk) * DSL + d] =
            part[kk][0][d] + part[kk][1][d] + part[kk][2][d] + part[kk][3][d];
    }
}

// ---------------------------------------------------------------------------
// Kernel 4: GRU cell, one block per (b,k) row, 192 threads (one per gate col).
// ---------------------------------------------------------------------------
__global__ __launch_bounds__(192) void sa_gru(
    const float* __restrict__ upd, float* __restrict__ slots,
    const float* __restrict__ W_ih, const float* __restrict__ W_hh,
    const float* __restrict__ b_ih, const float* __restrict__ b_hh)
{
    __shared__ float x[DSL], h[DSL], gx[3 * DSL], gh[3 * DSL];
    const int row = blockIdx.x, t = threadIdx.x;

    if (t < DSL) { x[t] = upd[row * DSL + t]; h[t] = slots[row * DSL + t]; }
    __syncthreads();

    float ax = b_ih[t], ah = b_hh[t];
    #pragma unroll 8
    for (int d = 0; d < DSL; ++d) {
        ax += x[d] * W_ih[d * 192 + t];   // weight loads coalesced over t
        ah += h[d] * W_hh[d * 192 + t];
    }
    gx[t] = ax; gh[t] = ah;
    __syncthreads();

    if (t < DSL) {
        float r = 1.f / (1.f + __expf(-(gx[t] + gh[t])));
        float z = 1.f / (1.f + __expf(-(gx[DSL + t] + gh[DSL + t])));
        float n = tanhf(gx[2 * DSL + t] + r * gh[2 * DSL + t]);
        slots[row * DSL + t] = (1.f - z) * n + z * h[t];
    }
}

// ---------------------------------------------------------------------------
// Kernel 5: validity head + copy slots to output. 1 block, 256 threads.
// ---------------------------------------------------------------------------
__global__ __launch_bounds__(256) void sa_final(
    const float* __restrict__ slots, const float* __restrict__ Wval,
    const float* __restrict__ bval, float* __restrict__ out_slots,
    float* __restrict__ out_mask)
{
    int row = threadIdx.x;
    float a = bval[0];
    #pragma unroll
    for (int d = 0; d < DSL; ++d) {
        float sv = slots[row * DSL + d];
        a += sv * Wval[d];
        out_slots[row * DSL + d] = sv;
    }
    out_mask[row] = (a > 0.f) ? 1.f : 0.f;
}

// ---------------------------------------------------------------------------
extern "C" void kernel_launch(void* const* d_in, const int* in_sizes, int n_in,
                              void* d_out, int out_size, void* d_ws, size_t ws_size,
                              hipStream_t stream) {
    const float* inputs  = (const float*)d_in[0];
    const int*   mask    = (const int*)  d_in[1];
    const float* slotemb = (const float*)d_in[2];
    const float* Wq      = (const float*)d_in[3];
    const float* Wk      = (const float*)d_in[4];
    const float* Wv      = (const float*)d_in[5];
    const float* W_ih    = (const float*)d_in[6];
    const float* W_hh    = (const float*)d_in[7];
    const float* b_ih    = (const float*)d_in[8];
    const float* b_hh    = (const float*)d_in[9];
    const float* ln_in_g = (const float*)d_in[10];
    const float* ln_in_b = (const float*)d_in[11];
    const float* ln_s_g  = (const float*)d_in[12];
    const float* ln_s_b  = (const float*)d_in[13];
    const float* Wval    = (const float*)d_in[14];
    const float* bval    = (const float*)d_in[15];

    // workspace layout (floats): k | v | slots | q | updates  (~67.3 MB)
    float* ws    = (float*)d_ws;
    float* kbuf  = ws;                               // 32*4096*64
    float* vbuf  = ws + (size_t)NB * SEQ * DSL;      // 32*4096*64
    float* slots = vbuf + (size_t)NB * SEQ * DSL;    // 32*8*64
    float* qbuf  = slots + NB * NSLOT * DSL;
    float* upd   = qbuf + NB * NSLOT * DSL;

    float* out       = (float*)d_out;
    float* out_slots = out;                          // 16384
    float* out_attn  = out + NB * NSLOT * DSL;       // 1048576
    float* out_mask  = out_attn + (size_t)NB * NSLOT * SEQ;  // 256

    sa_init_slots<<<NB * NSLOT * DSL / 256, 256, 0, stream>>>(slotemb, slots);
    sa_ln_kv<<<NB * SEQ / TILE_ROWS, 256, 0, stream>>>(
        inputs, Wk, Wv, ln_in_g, ln_in_b, kbuf, vbuf);

    for (int it = 0; it < 3; ++it) {
        sa_slot_q<<<1, 256, 0, stream>>>(slots, Wq, ln_s_g, ln_s_b, qbuf);
        sa_attn<<<NB, 256, 0, stream>>>(qbuf, kbuf, vbuf, mask, upd, out_attn,
                                        (it == 2) ? 1 : 0);
        sa_gru<<<NB * NSLOT, 192, 0, stream>>>(upd, slots, W_ih, W_hh, b_ih, b_hh);
    }
    sa_final<<<1, 256, 0, stream>>>(slots, Wval, bval, out_slots, out_mask);
}